// SwinTransformerBlock_87814901334736
// MI455X (gfx1250) — compile-verified
//
#include <hip/hip_runtime.h>
#include <math.h>

// ---------------------------------------------------------------------------
// Swin block for MI455X (gfx1250, wave32). All GEMMs via V_WMMA_F32_16X16X4_F32.
// ---------------------------------------------------------------------------

typedef float v2f __attribute__((ext_vector_type(2)));
typedef float v8f __attribute__((ext_vector_type(8)));

__device__ __forceinline__ v8f wmma_f32(float a0, float a1, float b0, float b1, v8f c) {
  v2f a; a.x = a0; a.y = a1;
  v2f b; b.x = b0; b.y = b1;
  // 8 args: (neg_a, A, neg_b, B, c_mod, C, reuse_a, reuse_b)
  return __builtin_amdgcn_wmma_f32_16x16x4_f32(false, a, false, b, (short)0, c, false, false);
}

__device__ __forceinline__ float wave_sum(float v) {
  v += __shfl_xor(v, 16, 32);
  v += __shfl_xor(v, 8, 32);
  v += __shfl_xor(v, 4, 32);
  v += __shfl_xor(v, 2, 32);
  v += __shfl_xor(v, 1, 32);
  return v;
}
__device__ __forceinline__ float wave_max(float v) {
  v = fmaxf(v, __shfl_xor(v, 16, 32));
  v = fmaxf(v, __shfl_xor(v, 8, 32));
  v = fmaxf(v, __shfl_xor(v, 4, 32));
  v = fmaxf(v, __shfl_xor(v, 2, 32));
  v = fmaxf(v, __shfl_xor(v, 1, 32));
  return v;
}

// Constants: DIM=128 NH=4 HD=32 WS=7 SS=3 H1=64 W1=32 H=70 W=35 NWIN=50 N=49
#define SCALE_QK 0.17677669529663687f

// ===========================================================================
// Kernel 1: one block per window. LN1 + shift + window gather -> QKV GEMM ->
// per-head QK^T -> bias+mask+softmax -> attn@V -> proj + shortcut residual.
// Writes x1[B,2048,128] (unpadded grid) to workspace.
// LDS: lds_x[64][128] (32KB) | lds_qkv[64][384] (96KB) | lds_at[4][64][64] (64KB)
// ===========================================================================
__global__ void __launch_bounds__(256) swin_attn_kernel(
    const float* __restrict__ xg,
    const float* __restrict__ n1g, const float* __restrict__ n1b,
    const float* __restrict__ qkvw, const float* __restrict__ qkvb,
    const float* __restrict__ rpb,
    const float* __restrict__ projw, const float* __restrict__ projb,
    float* __restrict__ x1)
{
  extern __shared__ float smem[];
  float* lds_x   = smem;                       // [64][128]
  float* lds_qkv = smem + 64 * 128;            // [64][384] q|k|v
  float* lds_at  = smem + 64 * 128 + 64 * 384; // [4][64][64]

  const int tid  = threadIdx.x;
  const int lane = tid & 31;
  const int wv   = tid >> 5;   // 8 waves
  const int ln15 = lane & 15;
  const int half = lane >> 4;

  const int blk  = blockIdx.x;
  const int bb   = blk / 50;
  const int wloc = blk % 50;
  const int wi   = wloc / 5;
  const int wj   = wloc % 5;

  // ---- Phase A: LN1 + cyclic shift + window partition (rows 49..63 zero) ----
  for (int r = wv * 8; r < wv * 8 + 8; ++r) {
    const int c = lane * 4;
    if (r < 49) {
      const int ti = r / 7, tj = r % 7;
      int h = wi * 7 + ti + 3; if (h >= 70) h -= 70;
      int w = wj * 7 + tj + 3; if (w >= 35) w -= 35;
      float4 v = make_float4(0.f, 0.f, 0.f, 0.f);
      if (h < 64 && w < 32) {
        const float4* p = (const float4*)(xg + ((size_t)(bb * 2048 + h * 32 + w)) * 128);
        v = p[lane];
      }
      const float mean = wave_sum(v.x + v.y + v.z + v.w) * (1.f / 128.f);
      const float sq   = wave_sum(v.x * v.x + v.y * v.y + v.z * v.z + v.w * v.w);
      const float var  = sq * (1.f / 128.f) - mean * mean;
      const float rstd = rsqrtf(var + 1e-5f);
      lds_x[r * 128 + c + 0] = (v.x - mean) * rstd * n1g[c + 0] + n1b[c + 0];
      lds_x[r * 128 + c + 1] = (v.y - mean) * rstd * n1g[c + 1] + n1b[c + 1];
      lds_x[r * 128 + c + 2] = (v.z - mean) * rstd * n1g[c + 2] + n1b[c + 2];
      lds_x[r * 128 + c + 3] = (v.w - mean) * rstd * n1g[c + 3] + n1b[c + 3];
    } else {
      lds_x[r * 128 + c + 0] = 0.f;
      lds_x[r * 128 + c + 1] = 0.f;
      lds_x[r * 128 + c + 2] = 0.f;
      lds_x[r * 128 + c + 3] = 0.f;
    }
  }
  __syncthreads();

  // ---- Phase B: QKV = xn @ qkv_w + b  (M=64, N=384, K=128), 96 tiles ----
  for (int tile = wv; tile < 96; tile += 8) {
    const int m0 = (tile / 24) * 16;
    const int n0 = (tile % 24) * 16;
    v8f acc = {};
    for (int k0 = 0; k0 < 128; k0 += 4) {
      const int kb = k0 + half * 2;
      const float a0 = lds_x[(m0 + ln15) * 128 + kb];
      const float a1 = lds_x[(m0 + ln15) * 128 + kb + 1];
      const float b0 = qkvw[kb * 384 + n0 + ln15];
      const float b1 = qkvw[(kb + 1) * 384 + n0 + ln15];
      acc = wmma_f32(a0, a1, b0, b1, acc);
    }
    const float bias = qkvb[n0 + ln15];
#pragma unroll
    for (int r = 0; r < 8; ++r)
      lds_qkv[(m0 + r + half * 8) * 384 + n0 + ln15] = acc[r] + bias;
  }
  __syncthreads();

  // ---- Phase C: logits = Q @ K^T per head, + rel-pos bias + shift mask ----
  for (int tile = wv; tile < 64; tile += 8) {
    const int hh = tile >> 4;
    const int m0 = ((tile >> 2) & 3) * 16;
    const int n0 = (tile & 3) * 16;
    v8f acc = {};
    for (int k0 = 0; k0 < 32; k0 += 4) {
      const int kb = k0 + half * 2;
      const float a0 = lds_qkv[(m0 + ln15) * 384 + hh * 32 + kb];
      const float a1 = lds_qkv[(m0 + ln15) * 384 + hh * 32 + kb + 1];
      const float b0 = lds_qkv[(n0 + ln15) * 384 + 128 + hh * 32 + kb];
      const float b1 = lds_qkv[(n0 + ln15) * 384 + 128 + hh * 32 + kb + 1];
      acc = wmma_f32(a0, a1, b0, b1, acc);
    }
#pragma unroll
    for (int r = 0; r < 8; ++r) {
      const int i = m0 + r + half * 8;
      const int j = n0 + ln15;
      float l = -1e30f;
      if (i < 49 && j < 49) {
        const int ri = i / 7, ci = i % 7, rj = j / 7, cj = j % 7;
        const int idx = (ri - rj + 6) * 13 + (ci - cj + 6);
        const int hsi = wi * 7 + ri, wsi = wj * 7 + ci;
        const int hsj = wi * 7 + rj, wsj = wj * 7 + cj;
        const int regi = (hsi < 63 ? 0 : (hsi < 67 ? 1 : 2)) * 3 + (wsi < 28 ? 0 : (wsi < 32 ? 1 : 2));
        const int regj = (hsj < 63 ? 0 : (hsj < 67 ? 1 : 2)) * 3 + (wsj < 28 ? 0 : (wsj < 32 ? 1 : 2));
        l = acc[r] * SCALE_QK + rpb[idx * 4 + hh] + (regi != regj ? -100.f : 0.f);
      }
      lds_at[hh * 4096 + i * 64 + j] = l;
    }
  }
  __syncthreads();

  // ---- Phase D: row softmax (4 heads x 64 rows; padded cols exp -> 0) ----
  for (int rr = 0; rr < 32; ++rr) {
    const int g  = wv * 32 + rr;
    const int hh = g >> 6;
    const int i  = g & 63;
    const float v0 = lds_at[hh * 4096 + i * 64 + lane];
    const float v1 = lds_at[hh * 4096 + i * 64 + lane + 32];
    float o0 = 0.f, o1 = 0.f;
    if (i < 49) {
      const float m  = wave_max(fmaxf(v0, v1));
      const float e0 = __expf(v0 - m);
      const float e1 = __expf(v1 - m);
      const float inv = 1.f / wave_sum(e0 + e1);
      o0 = e0 * inv; o1 = e1 * inv;
    }
    lds_at[hh * 4096 + i * 64 + lane]      = o0;
    lds_at[hh * 4096 + i * 64 + lane + 32] = o1;
  }
  __syncthreads();

  // ---- Phase E: out = attn @ V -> lds_x[64][128] (heads concatenated) ----
  for (int tile = wv; tile < 32; tile += 8) {
    const int hh = tile >> 3;
    const int m0 = ((tile >> 1) & 3) * 16;
    const int n0 = (tile & 1) * 16;
    v8f acc = {};
    for (int k0 = 0; k0 < 64; k0 += 4) {
      const int kb = k0 + half * 2;
      const float a0 = lds_at[hh * 4096 + (m0 + ln15) * 64 + kb];
      const float a1 = lds_at[hh * 4096 + (m0 + ln15) * 64 + kb + 1];
      const float b0 = lds_qkv[kb * 384 + 256 + hh * 32 + n0 + ln15];
      const float b1 = lds_qkv[(kb + 1) * 384 + 256 + hh * 32 + n0 + ln15];
      acc = wmma_f32(a0, a1, b0, b1, acc);
    }
#pragma unroll
    for (int r = 0; r < 8; ++r)
      lds_x[(m0 + r + half * 8) * 128 + hh * 32 + n0 + ln15] = acc[r];
  }
  __syncthreads();

  // ---- Phase F: proj + shortcut residual, scatter to unpadded grid ----
  for (int tile = wv; tile < 32; tile += 8) {
    const int m0 = (tile >> 3) * 16;
    const int n0 = (tile & 7) * 16;
    v8f acc = {};
    for (int k0 = 0; k0 < 128; k0 += 4) {
      const int kb = k0 + half * 2;
      const float a0 = lds_x[(m0 + ln15) * 128 + kb];
      const float a1 = lds_x[(m0 + ln15) * 128 + kb + 1];
      const float b0 = projw[kb * 128 + n0 + ln15];
      const float b1 = projw[(kb + 1) * 128 + n0 + ln15];
      acc = wmma_f32(a0, a1, b0, b1, acc);
    }
    const int c = n0 + ln15;
    const float pb = projb[c];
#pragma unroll
    for (int r = 0; r < 8; ++r) {
      const int t = m0 + r + half * 8;
      if (t < 49) {
        const int ti = t / 7, tj = t % 7;
        int h = wi * 7 + ti + 3; if (h >= 70) h -= 70;
        int w = wj * 7 + tj + 3; if (w >= 35) w -= 35;
        if (h < 64 && w < 32) {
          const size_t o = ((size_t)(bb * 2048 + h * 32 + w)) * 128 + c;
          x1[o] = xg[o] + acc[r] + pb;
        }
      }
    }
  }
}

// ===========================================================================
// Kernel 2: MLP. One block per 64 tokens. out = x1 + gelu(ln2(x1)@W1+b1)@W2+b2
// LDS: lds_x[64][128] (32KB) | lds_h[64][512] (128KB)
// ===========================================================================
__global__ void __launch_bounds__(256) swin_mlp_kernel(
    const float* __restrict__ x1,
    const float* __restrict__ n2g, const float* __restrict__ n2b,
    const float* __restrict__ w1, const float* __restrict__ b1v,
    const float* __restrict__ w2, const float* __restrict__ b2v,
    float* __restrict__ out)
{
  extern __shared__ float smem[];
  float* lds_x = smem;            // [64][128]
  float* lds_h = smem + 64 * 128; // [64][512]

  const int tid  = threadIdx.x;
  const int lane = tid & 31;
  const int wv   = tid >> 5;
  const int ln15 = lane & 15;
  const int half = lane >> 4;
  const size_t tb = (size_t)blockIdx.x * 64;

  // LN2
  for (int r = wv * 8; r < wv * 8 + 8; ++r) {
    const int c = lane * 4;
    const float4* p = (const float4*)(x1 + (tb + r) * 128);
    const float4 v = p[lane];
    const float mean = wave_sum(v.x + v.y + v.z + v.w) * (1.f / 128.f);
    const float sq   = wave_sum(v.x * v.x + v.y * v.y + v.z * v.z + v.w * v.w);
    const float rstd = rsqrtf(sq * (1.f / 128.f) - mean * mean + 1e-5f);
    lds_x[r * 128 + c + 0] = (v.x - mean) * rstd * n2g[c + 0] + n2b[c + 0];
    lds_x[r * 128 + c + 1] = (v.y - mean) * rstd * n2g[c + 1] + n2b[c + 1];
    lds_x[r * 128 + c + 2] = (v.z - mean) * rstd * n2g[c + 2] + n2b[c + 2];
    lds_x[r * 128 + c + 3] = (v.w - mean) * rstd * n2g[c + 3] + n2b[c + 3];
  }
  __syncthreads();

  // fc1 + exact GELU  (M=64, N=512, K=128), 128 tiles
  for (int tile = wv; tile < 128; tile += 8) {
    const int m0 = (tile >> 5) * 16;
    const int n0 = (tile & 31) * 16;
    v8f acc = {};
    for (int k0 = 0; k0 < 128; k0 += 4) {
      const int kb = k0 + half * 2;
      const float a0 = lds_x[(m0 + ln15) * 128 + kb];
      const float a1 = lds_x[(m0 + ln15) * 128 + kb + 1];
      const float b0 = w1[kb * 512 + n0 + ln15];
      const float b1 = w1[(kb + 1) * 512 + n0 + ln15];
      acc = wmma_f32(a0, a1, b0, b1, acc);
    }
    const float bias = b1v[n0 + ln15];
#pragma unroll
    for (int r = 0; r < 8; ++r) {
      float v = acc[r] + bias;
      v = 0.5f * v * (1.f + erff(v * 0.70710678118654752f));
      lds_h[(m0 + r + half * 8) * 512 + n0 + ln15] = v;
    }
  }
  __syncthreads();

  // fc2 + residual  (M=64, N=128, K=512), 32 tiles
  for (int tile = wv; tile < 32; tile += 8) {
    const int m0 = (tile >> 3) * 16;
    const int n0 = (tile & 7) * 16;
    v8f acc = {};
    for (int k0 = 0; k0 < 512; k0 += 4) {
      const int kb = k0 + half * 2;
      const float a0 = lds_h[(m0 + ln15) * 512 + kb];
      const float a1 = lds_h[(m0 + ln15) * 512 + kb + 1];
      const float b0 = w2[kb * 128 + n0 + ln15];
      const float b1 = w2[(kb + 1) * 128 + n0 + ln15];
      acc = wmma_f32(a0, a1, b0, b1, acc);
    }
    const int c = n0 + ln15;
    const float bias = b2v[c];
#pragma unroll
    for (int r = 0; r < 8; ++r) {
      const size_t o = (tb + m0 + r + half * 8) * 128 + c;
      out[o] = x1[o] + acc[r] + bias;
    }
  }
}

// ===========================================================================
extern "C" void kernel_launch(void* const* d_in, const int* in_sizes, int n_in,
                              void* d_out, int out_size, void* d_ws, size_t ws_size,
                              hipStream_t stream)
{
  (void)in_sizes; (void)n_in; (void)out_size; (void)ws_size;
  const float* x     = (const float*)d_in[0];
  const float* n1g   = (const float*)d_in[1];
  const float* n1b   = (const float*)d_in[2];
  const float* qkvw  = (const float*)d_in[3];
  const float* qkvb  = (const float*)d_in[4];
  const float* rpb   = (const float*)d_in[5];
  const float* projw = (const float*)d_in[6];
  const float* projb = (const float*)d_in[7];
  const float* n2g   = (const float*)d_in[8];
  const float* n2b   = (const float*)d_in[9];
  const float* fc1w  = (const float*)d_in[10];
  const float* fc1b  = (const float*)d_in[11];
  const float* fc2w  = (const float*)d_in[12];
  const float* fc2b  = (const float*)d_in[13];
  float* x1  = (float*)d_ws;          // [64, 2048, 128] f32 = 67 MB
  float* out = (float*)d_out;         // [64, 2048, 128] f32

  const int lds_attn = (64 * 128 + 64 * 384 + 4 * 64 * 64) * 4; // 196608 B
  const int lds_mlp  = (64 * 128 + 64 * 512) * 4;               // 163840 B
  hipFuncSetAttribute(reinterpret_cast<const void*>(swin_attn_kernel),
                      hipFuncAttributeMaxDynamicSharedMemorySize, lds_attn);
  hipFuncSetAttribute(reinterpret_cast<const void*>(swin_mlp_kernel),
                      hipFuncAttributeMaxDynamicSharedMemorySize, lds_mlp);

  swin_attn_kernel<<<64 * 50, 256, lds_attn, stream>>>(
      x, n1g, n1b, qkvw, qkvb, rpb, projw, projb, x1);
  swin_mlp_kernel<<<(64 * 2048) / 64, 256, lds_mlp, stream>>>(
      x1, n2g, n2b, fc1w, fc1b, fc2w, fc2b, out);
}